// FastAlphaLayer_59820304499462
// MI455X (gfx1250) — compile-verified
//
#include <hip/hip_runtime.h>
#include <math.h>

typedef __attribute__((ext_vector_type(2))) float v2f;
typedef __attribute__((ext_vector_type(8))) float v8f;

#define HW 256
#define NC 32
#define SAMPLE_ELEMS (HW * HW * NC)   // 2,097,152

// ---------------- Pass 1: per-block partial min/max ----------------
__global__ __launch_bounds__(256) void minmax_partial(const float* __restrict__ x,
                                                      float* __restrict__ pmin,
                                                      float* __restrict__ pmax) {
  __shared__ float smn[256], smx[256];
  const int b = blockIdx.x >> 6;       // 64 blocks per sample
  const int chunk = blockIdx.x & 63;
  const float4* p = reinterpret_cast<const float4*>(
      x + (size_t)b * SAMPLE_ELEMS + (size_t)chunk * 32768);
  float mn = 3.4e38f, mx = -3.4e38f;
  for (int i = threadIdx.x; i < 8192; i += 256) {
    float4 v = p[i];
    mn = fminf(mn, fminf(fminf(v.x, v.y), fminf(v.z, v.w)));
    mx = fmaxf(mx, fmaxf(fmaxf(v.x, v.y), fmaxf(v.z, v.w)));
  }
  smn[threadIdx.x] = mn;
  smx[threadIdx.x] = mx;
  __syncthreads();
  for (int s = 128; s > 0; s >>= 1) {
    if (threadIdx.x < s) {
      smn[threadIdx.x] = fminf(smn[threadIdx.x], smn[threadIdx.x + s]);
      smx[threadIdx.x] = fmaxf(smx[threadIdx.x], smx[threadIdx.x + s]);
    }
    __syncthreads();
  }
  if (threadIdx.x == 0) {
    pmin[blockIdx.x] = smn[0];
    pmax[blockIdx.x] = smx[0];
  }
}

// ---------------- Pass 2: per-sample final min/max ----------------
__global__ __launch_bounds__(64) void minmax_final(const float* __restrict__ pmin,
                                                   const float* __restrict__ pmax,
                                                   float* __restrict__ mn_out,
                                                   float* __restrict__ mx_out) {
  __shared__ float smn[64], smx[64];
  const int b = blockIdx.x;
  smn[threadIdx.x] = pmin[b * 64 + threadIdx.x];
  smx[threadIdx.x] = pmax[b * 64 + threadIdx.x];
  __syncthreads();
  for (int s = 32; s > 0; s >>= 1) {
    if (threadIdx.x < s) {
      smn[threadIdx.x] = fminf(smn[threadIdx.x], smn[threadIdx.x + s]);
      smx[threadIdx.x] = fmaxf(smx[threadIdx.x], smx[threadIdx.x + s]);
    }
    __syncthreads();
  }
  if (threadIdx.x == 0) {
    mn_out[b] = smn[0];
    mx_out[b] = smx[0];
  }
}

// Band matrices (0/1) for the vertical box-sum matmuls.
// S2[o] = sum_{p=o..o+15} r16[p]   (k=16: pad_begin=7, pad_end=8)
__device__ __forceinline__ float band16(int m, int k) {
  return (k >= m && k <= m + 15) ? 1.0f : 0.0f;
}
// S1[o] = sum_{q=o..o+3} r4[q]     (q = padded_row - 6; k=4: pb=1, pe=2)
__device__ __forceinline__ float band4(int m, int q) {
  return (q >= m && q <= m + 3) ? 1.0f : 0.0f;
}

// ---------------- Fused main kernel ----------------
// Shared pool layout (floats):
//   [0,8192)       buf : row batch buffer (31.0KB used) / output staging (32KB), reused
//   [8192,24560)   r16 : 31 rows x 16 n-slots (stride 33, bank-conflict-free) x 32 c
//   [24560,34592)  r4  : 19 rows x 16 n-slots (stride 33) x 32 c
//   [34592,34656)  sInv(32) + sBias(32)
#define SMEM_FLOATS 34656
#define OFF_R16 8192
#define OFF_R4 24560
#define OFF_PAR 34592
#define NSTRIDE 33   // padded n-slot stride: 33*n mod 64 bijective over n=0..15

__global__ __launch_bounds__(256) void fused_alpha_kernel(
    const float* __restrict__ x, const float* __restrict__ gamma,
    const float* __restrict__ beta, const float* __restrict__ mmean,
    const float* __restrict__ mvar, const float* __restrict__ mnmx,
    float* __restrict__ out) {
  __shared__ __align__(16) float smem[SMEM_FLOATS];
  float* buf = smem;
  float* r16 = smem + OFF_R16;
  float* r4 = smem + OFF_R4;
  float* sInv = smem + OFF_PAR;
  float* sBias = smem + OFF_PAR + 32;
  float4* buf4 = reinterpret_cast<float4*>(buf);

  const int tid = threadIdx.x;
  const int b = blockIdx.z;
  const int h0 = blockIdx.y * 16;
  const int w0 = blockIdx.x * 16;

  if (tid < 32) {
    float iv = gamma[tid] * rsqrtf(mvar[tid] + 1e-3f);
    sInv[tid] = iv;
    sBias[tid] = beta[tid] - mmean[tid] * iv;
  }
  const float mn = mnmx[b];
  const float mx = mnmx[16 + b];
  const float ir = 1.0f / (mx - mn);

  const float4* xb4 = reinterpret_cast<const float4*>(x + (size_t)b * SAMPLE_ELEMS);

  // Stream 31 padded rows (h0-7 .. h0+23) in 4 batches of <=8 rows.
  for (int batch = 0; batch < 4; ++batch) {
    const int p0 = batch * 8;
    const int nr = (batch == 3) ? 7 : 8;
    __syncthreads();  // buf reuse across batches
    const int tot4 = nr * 248;  // 31 w-positions * 8 float4 (32 ch) per row
    for (int idx = tid; idx < tot4; idx += 256) {
      const int pr = idx / 248;
      const int rem = idx - pr * 248;
      const int wl = rem >> 3;
      const int cq = rem & 7;
      const int h = h0 - 7 + p0 + pr;
      const int w = w0 - 7 + wl;
      float4 v = make_float4(0.0f, 0.0f, 0.0f, 0.0f);
      if ((unsigned)h < 256u && (unsigned)w < 256u) {
        v = xb4[((size_t)h * HW + (size_t)w) * 8 + cq];
        // Pull next batch's matching line toward the WGP (L2 is 192MB: input resident).
        if (h + 8 < 256)
          __builtin_prefetch(&xb4[((size_t)(h + 8) * HW + (size_t)w) * 8 + cq], 0, 0);
      }
      buf4[idx] = v;
    }
    __syncthreads();
    const int ch = tid;
    if (ch < nr * 32) {
      const int pr = ch >> 5;
      const int c = ch & 31;
      const float* row = buf + pr * 992 + c;  // stride 32 over w (conflict-free per wave)
      float P[31];
      float run = 0.0f;
#pragma unroll
      for (int w = 0; w < 31; ++w) {
        run += row[w * 32];
        P[w] = run;
      }
      const int p = p0 + pr;
      // r16[p][n][c]: horizontal 16-window sum at out col n (window = tile cols n..n+15)
      float* r16w = r16 + p * (16 * NSTRIDE) + c;
#pragma unroll
      for (int o = 0; o < 16; ++o)
        r16w[o * NSTRIDE] = P[o + 15] - (o ? P[o - 1] : 0.0f);
      // r4[p-6][n][c]: horizontal 4-window sum (window = tile cols n+6..n+9), rows 6..24 only
      if (p >= 6 && p <= 24) {
        float* r4w = r4 + (p - 6) * (16 * NSTRIDE) + c;
#pragma unroll
        for (int o = 0; o < 16; ++o)
          r4w[o * NSTRIDE] = P[o + 9] - P[o + 5];
      }
    }
  }
  __syncthreads();

  // ---- WMMA vertical band matmuls: each wave handles 4 channels ----
  const int lane = tid & 31;
  const int hh = (lane >> 4) & 1;  // half-wave selector
  const int m_lo = lane & 15;      // A-matrix M index
  const int n = lane & 15;         // B/D N index (out col)
  const int wave = tid >> 5;

  for (int ci = 0; ci < 4; ++ci) {
    const int c = wave * 4 + ci;

    // S2(16x16) = band16(16x31-pad-32) x r16(31x16), K chained in 8 chunks of 4
    v8f acc2 = {};
#pragma unroll
    for (int kc = 0; kc < 8; ++kc) {
      const int k0 = kc * 4 + 2 * hh;  // VGPR0 -> K=k0, VGPR1 -> K=k0+1
      v2f a, bm;
      a.x = band16(m_lo, k0);
      a.y = band16(m_lo, k0 + 1);
      bm.x = (k0 <= 30) ? r16[k0 * (16 * NSTRIDE) + n * NSTRIDE + c] : 0.0f;
      bm.y = (k0 + 1 <= 30) ? r16[(k0 + 1) * (16 * NSTRIDE) + n * NSTRIDE + c] : 0.0f;
      acc2 = __builtin_amdgcn_wmma_f32_16x16x4_f32(false, a, false, bm,
                                                   (short)0, acc2, false, false);
    }

    // S1(16x16) = band4(16x19-pad-20) x r4(19x16), K chained in 5 chunks of 4
    v8f acc1 = {};
#pragma unroll
    for (int kc = 0; kc < 5; ++kc) {
      const int k0 = kc * 4 + 2 * hh;
      v2f a, bm;
      a.x = band4(m_lo, k0);
      a.y = band4(m_lo, k0 + 1);
      bm.x = (k0 <= 18) ? r4[k0 * (16 * NSTRIDE) + n * NSTRIDE + c] : 0.0f;
      bm.y = (k0 + 1 <= 18) ? r4[(k0 + 1) * (16 * NSTRIDE) + n * NSTRIDE + c] : 0.0f;
      acc1 = __builtin_amdgcn_wmma_f32_16x16x4_f32(false, a, false, bm,
                                                   (short)0, acc1, false, false);
    }

    // Combine: count-corrected scaling, log-ratio (native log2), BN affine.
    const float iv = sInv[c];
    const float bs = sBias[c];
#pragma unroll
    for (int v = 0; v < 8; ++v) {
      const int m = v + 8 * hh;
      const int h = h0 + m;
      const int w = w0 + n;
      const float cnt2 = (float)((min(h + 8, 255) - max(h - 7, 0) + 1) *
                                 (min(w + 8, 255) - max(w - 7, 0) + 1));
      const float cnt1 = (float)((min(h + 2, 255) - max(h - 1, 0) + 1) *
                                 (min(w + 2, 255) - max(w - 1, 0) + 1));
      const float a2 = (acc2[v] - cnt2 * mn) * ir;  // boxsum16 of scaled xs
      const float a1 = (acc1[v] - cnt1 * mn) * ir;  // boxsum4  of scaled xs
      // (ln a2 - ln a1)/ln4 == (log2 a2 - log2 a1)/2 ; v_log_f32 is native log2
      const float alpha = 0.5f * (__log2f(a2) - __log2f(a1));
      buf[(m * 16 + n) * 32 + c] = alpha * iv + bs;
    }
  }
  __syncthreads();

  // ---- Coalesced vectorized tile writeout from staging ----
  float4* out4 = reinterpret_cast<float4*>(out);
  for (int idx = tid; idx < 2048; idx += 256) {
    const int m = idx >> 7;       // 128 float4 per output row chunk
    const int rem = idx & 127;
    const int nn = rem >> 3;
    const int cq = rem & 7;
    out4[(((size_t)(b * HW + h0 + m) * HW) + (size_t)(w0 + nn)) * 8 + cq] = buf4[idx];
  }
}

extern "C" void kernel_launch(void* const* d_in, const int* in_sizes, int n_in,
                              void* d_out, int out_size, void* d_ws, size_t ws_size,
                              hipStream_t stream) {
  const float* x = (const float*)d_in[0];
  const float* gamma = (const float*)d_in[1];
  const float* beta = (const float*)d_in[2];
  const float* mmean = (const float*)d_in[3];
  const float* mvar = (const float*)d_in[4];
  float* ws = (float*)d_ws;
  float* pmin = ws;          // 1024 floats
  float* pmax = ws + 1024;   // 1024 floats
  float* mnmx = ws + 2048;   // mn[16] then mx[16]

  minmax_partial<<<1024, 256, 0, stream>>>(x, pmin, pmax);
  minmax_final<<<16, 64, 0, stream>>>(pmin, pmax, mnmx, mnmx + 16);

  dim3 grid(HW / 16, HW / 16, 16);  // (w-tiles, h-tiles, samples)
  fused_alpha_kernel<<<grid, 256, 0, stream>>>(x, gamma, beta, mmean, mvar, mnmx,
                                               (float*)d_out);
}